// DigitCapsule_84542136254956
// MI455X (gfx1250) — compile-verified
//
#include <hip/hip_runtime.h>
#include <math.h>

// Problem constants (match reference)
#define B_ 64
#define R_ 2048
#define C_ 64
#define O_ 16
#define I_ 8
#define CO_ (C_ * O_)      // 1024
#define RI_ (R_ * I_)      // 16384
#define EPS_ 1e-8f

typedef __attribute__((ext_vector_type(2))) float v2f;
typedef __attribute__((ext_vector_type(8))) float v8f;

__device__ __forceinline__ v8f wmma_f32(v2f a, v2f b, v8f c) {
    return __builtin_amdgcn_wmma_f32_16x16x4_f32(false, a, false, b,
                                                 (short)0, c, false, false);
}

// -----------------------------------------------------------------------------
// xt[r][i][b] = x[b][r][i]  (one-time 4 MB transpose so the bbar GEMM's K dim
// (batch) is per-lane contiguous). Coalesced writes; reads are L2-absorbed.
// -----------------------------------------------------------------------------
__global__ __launch_bounds__(256) void caps_xt_kernel(
    const float* __restrict__ x, float* __restrict__ xt)
{
    const size_t t = (size_t)blockIdx.x * 256 + threadIdx.x; // 0 .. R*I*B-1
    const int b = (int)(t & 63);
    const size_t ri = t >> 6;
    const int i = (int)(ri & 7);
    const size_t r = ri >> 3;
    xt[t] = x[((size_t)b * R_ + r) * I_ + i];
}

// -----------------------------------------------------------------------------
// Pass S: v[b,c,o] = squash( sum_r coef(r,c) * sum_i W[r,c,o,i] * x[b,r,i] )
// Grid: (btile=4, c=64). Block: 256 = 8 waves, k-split over routes.
// WMMA: D(16x16: M=o, N=b) += A(16x4: W[*coef]) x B(4x16: x^T).
// K-permutation: step0 contracts i in {0,1,4,5}, step1 i in {2,3,6,7}
// (consistent for A and B) -> ONE b128 load per operand per route.
// Software pipelined: route r+1's operands are in flight during route r's WMMAs.
// -----------------------------------------------------------------------------
template <int UNIFORM>
__global__ __launch_bounds__(256) void caps_s_kernel(
    const float* __restrict__ x,    // [B, R, I]
    const float* __restrict__ W,    // [R, C, O, I]
    const float* __restrict__ cs,   // [C, R] route coefficients (unused if UNIFORM)
    float* __restrict__ v_out,      // [B, C, O]
    float* __restrict__ vt_out,     // [CO, B] transposed copy (for bbar)
    int write_vt)
{
    const int btile = blockIdx.x;          // 0..3
    const int c     = blockIdx.y;          // 0..63
    const int lane  = threadIdx.x & 31;
    const int widx  = threadIdx.x >> 5;    // 0..7 (k-split wave)
    const int l16   = lane & 15;
    const int half  = lane >> 4;

    v8f acc = {};

    const float* Wl  = W + (size_t)c * (O_ * I_) + (size_t)l16 * I_ + 4 * half;
    const float* xl  = x + (size_t)(btile * 16 + l16) * RI_ + 4 * half;
    const float* csl = cs + (size_t)c * R_;

    const int rbeg = widx * (R_ / 8);
    const int rend = rbeg + (R_ / 8);

    float4 wq = *(const float4*)(Wl + (size_t)rbeg * (CO_ * I_));
    float4 xq = *(const float4*)(xl + (size_t)rbeg * I_);
    float sc  = UNIFORM ? 0.0f : csl[rbeg];

    #pragma unroll 2
    for (int r = rbeg; r < rend; ++r) {
        const int rn = (r + 1 < rend) ? (r + 1) : r;
        const float4 wqn = *(const float4*)(Wl + (size_t)rn * (CO_ * I_));
        const float4 xqn = *(const float4*)(xl + (size_t)rn * I_);
        const float  scn = UNIFORM ? 0.0f : csl[rn];

        v2f a0, a1, b0, b1;
        if (UNIFORM) {
            a0.x = wq.x; a0.y = wq.y; a1.x = wq.z; a1.y = wq.w;
        } else {
            a0.x = wq.x * sc; a0.y = wq.y * sc;
            a1.x = wq.z * sc; a1.y = wq.w * sc;
        }
        b0.x = xq.x; b0.y = xq.y; b1.x = xq.z; b1.y = xq.w;
        acc = wmma_f32(a0, b0, acc);
        acc = wmma_f32(a1, b1, acc);

        wq = wqn; xq = xqn; sc = scn;
    }

    // Deterministic cross-wave reduction of the 8 partial 16x16 tiles via LDS.
    __shared__ float red[8 * 256];
    #pragma unroll
    for (int g = 0; g < 8; ++g) red[widx * 256 + g * 32 + lane] = acc[g];
    __syncthreads();

    const int t = threadIdx.x;
    float s = 0.0f;
    #pragma unroll
    for (int w = 0; w < 8; ++w) s += red[w * 256 + t];
    if (UNIFORM) s *= (1.0f / (float)R_);   // fold the uniform softmax weight here

    // slot -> (vgpr g, lane l): o = g + 8*(l>=16), b_local = l&15
    const int g = t >> 5;
    const int l = t & 31;
    const int o = g + ((l >> 4) << 3);
    const int b = btile * 16 + (l & 15);

    const float sq = s * s;
    const float vv = sq * s / ((1.0f + sq) * sqrtf(sq) + EPS_);
    v_out[(size_t)b * CO_ + (size_t)c * O_ + o] = vv;
    if (write_vt)
        vt_out[((size_t)c * O_ + o) * B_ + b] = vv;
}

// -----------------------------------------------------------------------------
// Pass BBAR: b_new[c][r] = b_old[c][r] + (1/B) sum_{i,o} W[r,c,o,i] * G_r[i,co]
// where G_r[i,co] = sum_b x[b,r,i] * v[b,co].
// One wave per route pair; A(16x64) preloaded; c-loop double-buffers the
// 8 b128 v-tile loads against the 16-WMMA chain. Per-c (i,o)-contraction
// partials are deferred to a wave-private LDS strip and reduced once at the
// end (replaces 4x ds_bpermute chains per c-tile with 1 ds_store).
// Grid: 128 blocks x 8 waves = 1024 route pairs. LDS: 64 KB/WG.
// -----------------------------------------------------------------------------
__global__ __launch_bounds__(256) void caps_bbar_kernel(
    const float* __restrict__ xt,    // [R, I, B]
    const float* __restrict__ W,     // [R, C, O, I]
    const float* __restrict__ vt,    // [CO, B]
    const float* __restrict__ b_old, // [C, R] (ignored if bold_zero)
    int bold_zero,
    float* __restrict__ b_new)       // [C, R]
{
    const int lane  = threadIdx.x & 31;
    const int widx  = threadIdx.x >> 5;
    const int wglob = blockIdx.x * 8 + widx;   // 0..1023
    const int r0    = 2 * wglob;
    const int l16   = lane & 15;
    const int half  = lane >> 4;
    const int rm    = r0 + (l16 >> 3);         // A row -> route
    const int im    = l16 & 7;                 //         -> i

    __shared__ float red[8 * 2048];            // per-wave 64c x 32lane strip
    float* myred = red + widx * 2048;

    // Preload A for K=64: step 2s uses (q.x,q.y), step 2s+1 uses (q.z,q.w);
    // step 2s contracts b in {8s,8s+1,8s+4,8s+5}, step 2s+1 the complement.
    v2f A[16];
    const float* xb = xt + ((size_t)rm * I_ + im) * B_ + 4 * half;
    #pragma unroll
    for (int s = 0; s < 8; ++s) {
        const float4 q = *(const float4*)(xb + 8 * s);
        A[2 * s].x     = q.x; A[2 * s].y     = q.y;
        A[2 * s + 1].x = q.z; A[2 * s + 1].y = q.w;
    }

    const int rsel = r0 + half;                // route this lane-half reduces
    const float* vl = vt + (size_t)l16 * B_ + 4 * half;            // + c*O_*B_
    const float* wl = W + ((size_t)rsel * (C_ * O_) + l16) * I_;   // + c*O_*I_

    float4 q[8];
    #pragma unroll
    for (int s = 0; s < 8; ++s) q[s] = *(const float4*)(vl + 8 * s);

    for (int c = 0; c < C_; ++c) {
        const int cn = (c + 1 < C_) ? (c + 1) : c;
        float4 qn[8];
        #pragma unroll
        for (int s = 0; s < 8; ++s)
            qn[s] = *(const float4*)(vl + (size_t)cn * (O_ * B_) + 8 * s);
        const float4 w0 = *(const float4*)(wl + (size_t)c * (O_ * I_));
        const float4 w1 = *(const float4*)(wl + (size_t)c * (O_ * I_) + 4);

        v8f acc = {};
        #pragma unroll
        for (int s = 0; s < 8; ++s) {
            v2f b0, b1;
            b0.x = q[s].x; b0.y = q[s].y;
            b1.x = q[s].z; b1.y = q[s].w;
            acc = wmma_f32(A[2 * s],     b0, acc);
            acc = wmma_f32(A[2 * s + 1], b1, acc);
        }
        // acc[g] = G[M = g + 8*half][N = l16]; contract i=g with W[rsel,c,o=l16,i]
        const float p = acc[0] * w0.x + acc[1] * w0.y + acc[2] * w0.z + acc[3] * w0.w
                      + acc[4] * w1.x + acc[5] * w1.y + acc[6] * w1.z + acc[7] * w1.w;
        myred[c * 32 + lane] = p;              // deferred o-reduction

        #pragma unroll
        for (int s = 0; s < 8; ++s) q[s] = qn[s];
    }

    // Same-wave DS ops are in-order; wait for store data then reduce.
    asm volatile("s_wait_dscnt 0x0" ::: "memory");

    // lane -> rsub = half, c = (l16*4 + k): sum 16 o-partials per output.
    const int rg = r0 + half;
    #pragma unroll
    for (int k = 0; k < 4; ++k) {
        const int c = (l16 << 2) + k;
        const float* rr = myred + c * 32 + half * 16;
        const float4 u0 = *(const float4*)(rr);
        const float4 u1 = *(const float4*)(rr + 4);
        const float4 u2 = *(const float4*)(rr + 8);
        const float4 u3 = *(const float4*)(rr + 12);
        const float sum = ((u0.x + u0.y) + (u0.z + u0.w))
                        + ((u1.x + u1.y) + (u1.z + u1.w))
                        + ((u2.x + u2.y) + (u2.z + u2.w))
                        + ((u3.x + u3.y) + (u3.z + u3.w));
        const float prev = bold_zero ? 0.0f : b_old[(size_t)c * R_ + rg];
        b_new[(size_t)c * R_ + rg] = prev + sum * (1.0f / (float)B_);
    }
}

// -----------------------------------------------------------------------------
// Softmax over routes (axis=0 of b_ij); storage is [C][R] -> contiguous rows.
// -----------------------------------------------------------------------------
__global__ __launch_bounds__(256) void caps_softmax_kernel(
    const float* __restrict__ b,   // [C, R]
    float* __restrict__ cs)        // [C, R]
{
    const int c = blockIdx.x;
    const int t = threadIdx.x;
    const float* row = b + (size_t)c * R_;
    float* out = cs + (size_t)c * R_;

    __shared__ float sm[256];

    float m = -INFINITY;
    for (int r = t; r < R_; r += 256) m = fmaxf(m, row[r]);
    sm[t] = m;
    __syncthreads();
    for (int s = 128; s > 0; s >>= 1) {
        if (t < s) sm[t] = fmaxf(sm[t], sm[t + s]);
        __syncthreads();
    }
    const float mx = sm[0];
    __syncthreads();

    float sum = 0.0f;
    for (int r = t; r < R_; r += 256) sum += expf(row[r] - mx);
    sm[t] = sum;
    __syncthreads();
    for (int s = 128; s > 0; s >>= 1) {
        if (t < s) sm[t] += sm[t + s];
        __syncthreads();
    }
    const float inv = 1.0f / sm[0];

    for (int r = t; r < R_; r += 256) out[r] = expf(row[r] - mx) * inv;
}

// -----------------------------------------------------------------------------
// Host launcher: 3 routing iterations, u_hat (537 MB) never materialized.
// Workspace (floats): xt 1048576 | v0 65536 | vt 65536 | b1/cs/b2 131072 each.
// Total ~6.2 MB.
// -----------------------------------------------------------------------------
extern "C" void kernel_launch(void* const* d_in, const int* in_sizes, int n_in,
                              void* d_out, int out_size, void* d_ws, size_t ws_size,
                              hipStream_t stream) {
    (void)in_sizes; (void)n_in; (void)out_size; (void)ws_size;
    const float* x = (const float*)d_in[0];   // [B,R,I]
    const float* W = (const float*)d_in[1];   // [R,C,O,I]
    float* out = (float*)d_out;               // [B,C,O,1] == [B,C,O]

    float* ws = (float*)d_ws;
    float* xt = ws;                            // R*I*B = 1048576
    float* v0 = xt + (size_t)R_ * I_ * B_;     // B*CO  = 65536
    float* vt = v0 + (size_t)B_ * CO_;         // CO*B  = 65536
    float* b1 = vt + (size_t)CO_ * B_;         // C*R   = 131072
    float* cs = b1 + (size_t)C_ * R_;          // C*R   = 131072
    float* b2 = cs + (size_t)C_ * R_;          // C*R   = 131072

    dim3 gs(4, C_), bs(256);

    caps_xt_kernel<<<(R_ * I_ * B_) / 256, 256, 0, stream>>>(x, xt);

    // it = 0: softmax(0) is uniform 1/R -> fused (scale applied at epilogue)
    caps_s_kernel<1><<<gs, bs, 0, stream>>>(x, W, cs, v0, vt, 1);
    caps_bbar_kernel<<<128, 256, 0, stream>>>(xt, W, vt, b1, 1, b1);

    // it = 1
    caps_softmax_kernel<<<C_, 256, 0, stream>>>(b1, cs);
    caps_s_kernel<0><<<gs, bs, 0, stream>>>(x, W, cs, v0, vt, 1);
    caps_bbar_kernel<<<128, 256, 0, stream>>>(xt, W, vt, b1, 0, b2);

    // it = 2 (final: write v straight to d_out, no vt needed)
    caps_softmax_kernel<<<C_, 256, 0, stream>>>(b2, cs);
    caps_s_kernel<0><<<gs, bs, 0, stream>>>(x, W, cs, out, vt, 0);
}